// SimpleEdgeScorer_34016140985093
// MI455X (gfx1250) — compile-verified
//
#include <hip/hip_runtime.h>

// ---------------------------------------------------------------------------
// SimpleEdgeScorer on MI455X (gfx1250, wave32, WMMA).
//
// Roofline: ~9 GFLOP fp32 total vs ~1.9 GB of L2-resident gather/scatter
// traffic -> memory bound. Use exact-precision V_WMMA_F32_16X16X4_F32 for all
// dense matmuls (no precision downgrade needed; matrix pipe is nowhere near
// the bottleneck), atomics for the segment sums.
// ---------------------------------------------------------------------------

#define H 64

typedef __attribute__((ext_vector_type(2))) float v2f;
typedef __attribute__((ext_vector_type(8))) float v8f;

__global__ void zero_f32(float* __restrict__ p, long n) {
    long i = (long)blockIdx.x * blockDim.x + threadIdx.x;
    if (i < n) p[i] = 0.0f;
}

// x[n, f] = embed[type[n], f]
__global__ void gather_embed(const int* __restrict__ types,
                             const float* __restrict__ embed,
                             float* __restrict__ x, int n_nodes) {
    int i = blockIdx.x * blockDim.x + threadIdx.x;
    if (i >= n_nodes * H) return;
    int node = i >> 6;
    int f    = i & (H - 1);
    x[i] = embed[types[node] * H + f];
}

// deg[dst[e]] += 1
__global__ void degree_count(const int* __restrict__ dst,
                             float* __restrict__ deg, int n_edges) {
    int e = blockIdx.x * blockDim.x + threadIdx.x;
    if (e < n_edges) atomicAdd(&deg[dst[e]], 1.0f);
}

// agg[dst[e], f] += x[src[e], f]   (one thread per edge-feature)
__global__ void scatter_feat(const float* __restrict__ x,
                             const int* __restrict__ src,
                             const int* __restrict__ dst,
                             float* __restrict__ agg, long n_ef) {
    long gid = (long)blockIdx.x * blockDim.x + threadIdx.x;
    if (gid >= n_ef) return;
    long e = gid >> 6;
    int  f = (int)(gid & (H - 1));
    atomicAdd(&agg[(long)dst[e] * H + f], x[(long)src[e] * H + f]);
}

// One wave per 16-node tile: xout = relu((x + agg/max(deg,1)) @ W + b), K=64.
// In-place safe: each wave only touches its own 16 rows.
__global__ void __launch_bounds__(32)
gnn_layer_wmma(const float* x, const float* __restrict__ agg,
               const float* __restrict__ deg, const float* __restrict__ W,
               const float* __restrict__ bias, float* xout) {
    const int tile = blockIdx.x;
    const int lane = threadIdx.x & 31;
    const int m    = lane & 15;           // A row / B col / C col index
    const int kb   = (lane >> 4) * 2;     // K sub-pair selected by lane half
    const int node = tile * 16 + m;

    const float invd = 1.0f / fmaxf(deg[node], 1.0f);
    const float* xr = x   + (long)node * H;
    const float* ar = agg + (long)node * H;

    // A fragments for the full K=64 sweep (16 steps of K=4).
    v2f afrag[16];
#pragma unroll
    for (int kk = 0; kk < 16; ++kk) {
        int k = kk * 4 + kb;
        v2f xv = *(const v2f*)(xr + k);
        v2f av = *(const v2f*)(ar + k);
        afrag[kk].x = xv.x + av.x * invd;
        afrag[kk].y = xv.y + av.y * invd;
    }

#pragma unroll
    for (int nt = 0; nt < 4; ++nt) {
        const int col = nt * 16 + m;
        v8f c = {};
#pragma unroll
        for (int kk = 0; kk < 16; ++kk) {
            int k = kk * 4 + kb;
            v2f bf;
            bf.x = W[(k + 0) * H + col];
            bf.y = W[(k + 1) * H + col];
            c = __builtin_amdgcn_wmma_f32_16x16x4_f32(
                    false, afrag[kk], false, bf, (short)0, c, false, false);
        }
        const float bv = bias[col];
#pragma unroll
        for (int r = 0; r < 8; ++r) {
            int row = r + (lane >> 4) * 8;
            xout[(long)(tile * 16 + row) * H + col] = fmaxf(c[r] + bv, 0.0f);
        }
    }
}

// One wave per 16-query-edge tile:
//   h = concat(x[src], x[dst])  (K=128)
//   y1 = relu(h @ We1 + be1)    -> 16x64 tile via WMMA, staged to LDS
//   out = y1 @ We2 + be2        -> 64-term dot per edge
__global__ void __launch_bounds__(64)
edge_mlp_wmma(const float* __restrict__ x, const int* __restrict__ qsrc,
              const int* __restrict__ qdst, const float* __restrict__ We1,
              const float* __restrict__ be1, const float* __restrict__ We2,
              const float* __restrict__ be2, float* __restrict__ out) {
    __shared__ float lds[2][16 * H];

    const int wave = threadIdx.x >> 5;
    const int lane = threadIdx.x & 31;
    const int tile = blockIdx.x * 2 + wave;
    const int m    = lane & 15;
    const int kb   = (lane >> 4) * 2;
    const int edge = tile * 16 + m;

    const long s = qsrc[edge];
    const long d = qdst[edge];

    // A fragments, K = 0..127: first 64 from src row, next 64 from dst row.
    v2f afrag[32];
#pragma unroll
    for (int kk = 0; kk < 16; ++kk)
        afrag[kk] = *(const v2f*)(x + s * H + kk * 4 + kb);
#pragma unroll
    for (int kk = 0; kk < 16; ++kk)
        afrag[16 + kk] = *(const v2f*)(x + d * H + kk * 4 + kb);

#pragma unroll
    for (int nt = 0; nt < 4; ++nt) {
        const int col = nt * 16 + m;
        v8f c = {};
#pragma unroll
        for (int kk = 0; kk < 32; ++kk) {
            int k = kk * 4 + kb;
            v2f bf;
            bf.x = We1[(k + 0) * H + col];
            bf.y = We1[(k + 1) * H + col];
            c = __builtin_amdgcn_wmma_f32_16x16x4_f32(
                    false, afrag[kk], false, bf, (short)0, c, false, false);
        }
        const float bv = be1[col];
#pragma unroll
        for (int r = 0; r < 8; ++r) {
            int row = r + (lane >> 4) * 8;
            lds[wave][row * H + col] = fmaxf(c[r] + bv, 0.0f);
        }
    }

    __syncthreads();

    if (lane < 16) {
        float acc = be2[0];
        const float* yr = &lds[wave][lane * H];
#pragma unroll
        for (int cidx = 0; cidx < H; ++cidx) acc += yr[cidx] * We2[cidx];
        out[tile * 16 + lane] = acc;
    }
}

extern "C" void kernel_launch(void* const* d_in, const int* in_sizes, int n_in,
                              void* d_out, int out_size, void* d_ws, size_t ws_size,
                              hipStream_t stream) {
    const int*   types = (const int*)d_in[0];
    const int*   gsrc  = (const int*)d_in[1];
    const int*   gdst  = (const int*)d_in[2];
    const int*   qsrc  = (const int*)d_in[3];
    const int*   qdst  = (const int*)d_in[4];
    const float* embed = (const float*)d_in[5];
    const float* W1    = (const float*)d_in[6];
    const float* b1    = (const float*)d_in[7];
    const float* W2    = (const float*)d_in[8];
    const float* b2    = (const float*)d_in[9];
    const float* We1   = (const float*)d_in[10];
    const float* be1   = (const float*)d_in[11];
    const float* We2   = (const float*)d_in[12];
    const float* be2   = (const float*)d_in[13];

    const int n_nodes = in_sizes[0];   // 50000 (divisible by 16)
    const int n_edges = in_sizes[1];   // 1.6M
    const int n_query = in_sizes[3];   // 500000 (divisible by 32)

    // Workspace layout: x[N*64] | agg[N*64] | deg[N]  (~26 MB)
    float* x   = (float*)d_ws;
    float* agg = x + (size_t)n_nodes * H;
    float* deg = agg + (size_t)n_nodes * H;
    float* outf = (float*)d_out;

    const long nx  = (long)n_nodes * H;      // node-feature count
    const long nef = (long)n_edges * H;      // edge-feature count
    const int  B   = 256;

    // Type-embedding gather + degree counts (deg fixed across layers).
    zero_f32<<<(unsigned)((n_nodes + B - 1) / B), B, 0, stream>>>(deg, n_nodes);
    gather_embed<<<(unsigned)((nx + B - 1) / B), B, 0, stream>>>(types, embed, x, n_nodes);
    degree_count<<<(unsigned)((n_edges + B - 1) / B), B, 0, stream>>>(gdst, deg, n_edges);

    const unsigned gnn_blocks = (unsigned)(n_nodes / 16);      // 1 wave per 16-row tile

    // Layer 1
    zero_f32<<<(unsigned)((nx + B - 1) / B), B, 0, stream>>>(agg, nx);
    scatter_feat<<<(unsigned)((nef + B - 1) / B), B, 0, stream>>>(x, gsrc, gdst, agg, nef);
    gnn_layer_wmma<<<gnn_blocks, 32, 0, stream>>>(x, agg, deg, W1, b1, x);

    // Layer 2
    zero_f32<<<(unsigned)((nx + B - 1) / B), B, 0, stream>>>(agg, nx);
    scatter_feat<<<(unsigned)((nef + B - 1) / B), B, 0, stream>>>(x, gsrc, gdst, agg, nef);
    gnn_layer_wmma<<<gnn_blocks, 32, 0, stream>>>(x, agg, deg, W2, b2, x);

    // Edge MLP over query edges: 2 waves (tiles) per block.
    const unsigned q_blocks = (unsigned)(n_query / 32);
    edge_mlp_wmma<<<q_blocks, 64, 0, stream>>>(x, qsrc, qdst, We1, be1, We2, be2, outf);
}